// TransPhormer_72808285602163
// MI455X (gfx1250) — compile-verified
//
#include <hip/hip_runtime.h>
#include <hip/hip_bf16.h>
#include <math.h>

// ---------------- constants (match reference) ----------------
#define N_NODES 4096
#define N_EDGES 32768
#define NBASIS  16
#define NHEADS  8
#define NIRR    112
#define NDIM    240
#define EC      4096          // edge chunk (keeps coup+kv chunks L2-resident)
#define NCHUNK  (N_EDGES / EC)
#define SCALE_F 0.094491118252306805f   // 1/sqrt(112)

typedef float v2f __attribute__((ext_vector_type(2)));
typedef float v8f __attribute__((ext_vector_type(8)));

#if defined(__has_builtin)
#if __has_builtin(__builtin_amdgcn_global_load_async_to_lds_b32) && \
    __has_builtin(__builtin_amdgcn_s_wait_asynccnt)
#define HAVE_ASYNC_LDS 1
#endif
#endif

typedef __attribute__((address_space(1))) int as1_int;
typedef __attribute__((address_space(3))) int as3_int;

// irrep geometry
constexpr int MUL_[3]  = {64, 32, 16};
constexpr int DIM_[3]  = {1, 3, 5};
constexpr int OFFL_[3] = {0, 64, 160};     // node-block float offsets (stride 240)
constexpr int QOFF_[3] = {0, 512, 1280};   // q / kv per-row block offsets (stride 1920)
constexpr int B3_[3]   = {0, 112, 544};    // coup per-edge block offsets (stride 1184)
constexpr int CM_[3]   = {112, 144, 128};  // coupled multiplicities
constexpr int RSOFF_[3]= {0, 1, 4};        // rsh slices

// CG paths (l1,l2,l3) in reference order
constexpr int P_L1[11] = {0,1,2, 0,1,1,2, 0,1,2,2};
constexpr int P_L2[11] = {0,1,2, 1,0,2,1, 2,1,0,2};
constexpr int P_L3[11] = {0,0,0, 1,1,1,1, 2,2,2,2};
constexpr int P_M [11] = {64,32,16, 64,32,32,16, 64,32,16,16};   // mul of path = MUL[l1]
constexpr int P_UO[11] = {0,64,96, 0,64,96,128, 0,64,96,112};    // offset inside coup[l3]
constexpr int CG_OFFP[11] = {0,1,10,35,44,53,98,143,168,213,238}; // flat CG table offsets
// sizes: 1,9,25,9,9,45,45,25,45,25,125 -> total 363

// ---------------- workspace layout (float elements) ----------------
constexpr long CG_WS   = 0;                              // 363 (padded to 384)
constexpr long ESC_WS  = 384;                            // E*3
constexpr long SRCF_WS = ESC_WS  + 3L  * N_EDGES;        // N*240
constexpr long DSTF_WS = SRCF_WS + 240L * N_NODES;
constexpr long Q_WS    = DSTF_WS + 240L * N_NODES;       // N*1920
constexpr long LOG_WS  = Q_WS    + 1920L * N_NODES;      // E*8  (logits -> exp)
constexpr long SMAX_WS = LOG_WS  + 8L * N_EDGES;         // N*8 (as uint)
constexpr long SSUM_WS = SMAX_WS + 8L * N_NODES;         // N*8
constexpr long MSG_WS  = SSUM_WS + 8L * N_NODES;         // N*1920
constexpr long MB_WS   = MSG_WS  + 1920L * N_NODES;      // N*240
constexpr long X0_WS   = MB_WS   + 240L * N_NODES;       // N*112
constexpr long X1N_WS  = X0_WS   + 112L * N_NODES;
constexpr long H1_WS   = X1N_WS  + 112L * N_NODES;
constexpr long H2_WS   = H1_WS   + 112L * N_NODES;
constexpr long COUP_WS = H2_WS   + 112L * N_NODES;       // EC*1184
constexpr long KV_WS   = COUP_WS + 1184L * EC;           // EC*1920
// total ~ 33.7M floats ~ 135 MB

// =====================================================================
// 0) on-device Clebsch-Gordan table generation (deterministic each call)
// =====================================================================
__device__ __forceinline__ double dfact(int n) {
  double r = 1.0; for (int i = 2; i <= n; ++i) r *= (double)i; return r;
}

__device__ double su2cg(int j1, int j2, int j3, int m1, int m2) {
  int m3 = m1 + m2;
  if (m3 < -j3 || m3 > j3) return 0.0;
  double pref = sqrt((2.0*j3+1.0)*dfact(j3+j1-j2)*dfact(j3-j1+j2)*dfact(j1+j2-j3)/dfact(j1+j2+j3+1));
  pref *= sqrt(dfact(j3+m3)*dfact(j3-m3)*dfact(j1-m1)*dfact(j1+m1)*dfact(j2-m2)*dfact(j2+m2));
  int k0 = 0;
  if (j2-j3-m1 > k0) k0 = j2-j3-m1;
  if (j1-j3+m2 > k0) k0 = j1-j3+m2;
  int k1 = j1+j2-j3;
  if (j1-m1 < k1) k1 = j1-m1;
  if (j2+m2 < k1) k1 = j2+m2;
  double s = 0.0;
  for (int k = k0; k <= k1; ++k) {
    double t = dfact(k)*dfact(j1+j2-j3-k)*dfact(j1-m1-k)*dfact(j2+m2-k)*dfact(j3-j2+m1+k)*dfact(j3-j1-m2+k);
    s += ((k & 1) ? -1.0 : 1.0) / t;
  }
  return pref * s;
}

__device__ void build_q(int l, double qr[5][5], double qi[5][5]) {
  for (int a = 0; a < 5; ++a)
    for (int b = 0; b < 5; ++b) { qr[a][b] = 0.0; qi[a][b] = 0.0; }
  const double is2 = 0.70710678118654752440;
  for (int m = -l; m < 0; ++m) {
    qr[l+m][l-m] = is2;        // 1/sqrt2 at col l+|m|
    qi[l+m][l+m] = -is2;       // -i/sqrt2 at col l-|m|
  }
  qr[l][l] = 1.0;
  for (int m = 1; m <= l; ++m) {
    double sgn = (m & 1) ? -1.0 : 1.0;
    qr[l+m][l+m] = sgn * is2;
    qi[l+m][l-m] = sgn * is2;
  }
  // phase (-i)^l
  double pre, pim;
  if (l == 0)      { pre = 1.0;  pim = 0.0; }
  else if (l == 1) { pre = 0.0;  pim = -1.0; }
  else             { pre = -1.0; pim = 0.0; }
  int d = 2*l + 1;
  for (int a = 0; a < d; ++a)
    for (int b = 0; b < d; ++b) {
      double re = qr[a][b], im = qi[a][b];
      qr[a][b] = re*pre - im*pim;
      qi[a][b] = re*pim + im*pre;
    }
}

__global__ void cg_gen_kernel(float* __restrict__ cg) {
  if (threadIdx.x != 0 || blockIdx.x != 0) return;
  for (int p = 0; p < 11; ++p) {
    int l1 = P_L1[p], l2 = P_L2[p], l3 = P_L3[p];
    int d1 = 2*l1+1, d2 = 2*l2+1, d3 = 2*l3+1;
    double q1r[5][5], q1i[5][5], q2r[5][5], q2i[5][5], q3r[5][5], q3i[5][5];
    build_q(l1, q1r, q1i);
    build_q(l2, q2r, q2i);
    build_q(l3, q3r, q3i);
    for (int a = 0; a < d1; ++a)
      for (int b = 0; b < d2; ++b)
        for (int c = 0; c < d3; ++c) {
          double sre = 0.0;
          for (int i = 0; i < d1; ++i)
            for (int j = 0; j < d2; ++j) {
              int m1 = i - l1, m2 = j - l2;
              int k = l3 + m1 + m2;
              if (k < 0 || k >= d3) continue;
              double s = su2cg(l1, l2, l3, m1, m2);
              if (s == 0.0) continue;
              double tr = q1r[a][i]*q2r[b][j] - q1i[a][i]*q2i[b][j];
              double ti = q1r[a][i]*q2i[b][j] + q1i[a][i]*q2r[b][j];
              double cr = q3r[c][k], ci = -q3i[c][k];
              sre += (tr*cr - ti*ci) * s;
            }
          cg[CG_OFFP[p] + (a*d2 + b)*d3 + c] = (float)sre;
        }
  }
}

// =====================================================================
// 1) strided fp32 WMMA GEMM with LDS-staged B tile:
//    C[M,N] = A[M,K] * B[K,N]  (B row-major, unit column stride)
//    block = 128 threads = 4 waves x 4 M-tiles; each wave owns 2 N-tiles.
//    B tile (K x 32) staged to LDS via GLOBAL_LOAD_ASYNC_TO_LDS (if avail),
//    consumed by ds loads; K unrolled x2 -> 4 v_wmma per iteration.
// =====================================================================
template <bool AK1>
__global__ void wmma_gemm_f32_t(const float* __restrict__ A, long sAm, long sAk,
                                const float* __restrict__ B, long sBk,
                                float* __restrict__ C, long sCm, long sCn,
                                int K, int Ntot) {
  extern __shared__ float bs[];       // K x 32 tile of B
  const int tid  = threadIdx.x;
  const int lane = tid & 31;
  const int wave = tid >> 5;
  const int half = lane >> 4;         // 0: K=k,k+1   1: K=k+2,k+3
  const int l16  = lane & 15;
  const int n0   = blockIdx.y * 32;

  // --- stage B tile into LDS (async DMA path on CDNA5) ---
  for (int idx = tid; idx < K * 32; idx += 128) {
    const int kk = idx >> 5;
    const int nn = n0 + (idx & 31);
    if (nn < Ntot) {
#ifdef HAVE_ASYNC_LDS
      __builtin_amdgcn_global_load_async_to_lds_b32(
          (as1_int*)(uintptr_t)(B + (long)kk * sBk + nn),
          (as3_int*)(uintptr_t)(bs + idx), 0, 0);
#else
      bs[idx] = B[(long)kk * sBk + nn];
#endif
    } else {
      bs[idx] = 0.0f;
    }
  }
#ifdef HAVE_ASYNC_LDS
  __builtin_amdgcn_s_wait_asynccnt(0);
#endif
  __syncthreads();

  const long mt = (long)blockIdx.x * 4 + wave;
  const float* Ap = A + (mt * 16 + l16) * sAm;
  v8f acc0 = {}, acc1 = {};
  for (int k = 0; k < K; k += 8) {
#pragma unroll
    for (int kk = 0; kk < 8; kk += 4) {
      const int ka = k + kk + half * 2;
      v2f a, b0, b1;
      if (AK1) {
        a = *(const v2f*)(Ap + ka);                       // contiguous A pair (b64)
      } else {
        a.x = Ap[(long)ka * sAk];
        a.y = Ap[(long)(ka + 1) * sAk];
      }
      b0.x = bs[ka * 32 + l16];
      b0.y = bs[(ka + 1) * 32 + l16];
      b1.x = bs[ka * 32 + 16 + l16];
      b1.y = bs[(ka + 1) * 32 + 16 + l16];
      acc0 = __builtin_amdgcn_wmma_f32_16x16x4_f32(false, a, false, b0, (short)0, acc0, false, false);
      acc1 = __builtin_amdgcn_wmma_f32_16x16x4_f32(false, a, false, b1, (short)0, acc1, false, false);
    }
    if (!AK1 && k + 16 < K)
      __builtin_prefetch(Ap + (long)(k + 16) * sAk, 0, 1);
  }
  // C layout: VGPR r -> (M = half*8 + r, N = l16)
  const long m0 = mt * 16 + half * 8;
  const int c0 = n0 + l16, c1 = n0 + 16 + l16;
  float* Cp0 = C + (long)c0 * sCn;
  float* Cp1 = C + (long)c1 * sCn;
#pragma unroll
  for (int r = 0; r < 8; ++r) {
    Cp0[(m0 + r) * sCm] = acc0[r];
    if (c1 < Ntot) Cp1[(m0 + r) * sCm] = acc1[r];
  }
}

// =====================================================================
// 2) small elementwise / per-edge kernels
// =====================================================================
__global__ void init_kernel(float* __restrict__ msg, unsigned* __restrict__ smax,
                            float* __restrict__ ssum) {
  long i = (long)blockIdx.x * blockDim.x + threadIdx.x;
  if (i < (long)N_NODES * 1920) msg[i] = 0.0f;
  if (i < (long)N_NODES * 8) { smax[i] = 0u; ssum[i] = 0.0f; }
}

__global__ void escal_kernel(const float* __restrict__ rbf, const float* __restrict__ Wrbf,
                             const float* __restrict__ brbf, float* __restrict__ escal) {
  int idx = blockIdx.x * blockDim.x + threadIdx.x;
  if (idx >= N_EDGES * 3) return;
  int e = idx / 3, c = idx % 3;
  float acc = brbf[c];
#pragma unroll
  for (int b = 0; b < NBASIS; ++b) acc += rbf[(long)e * NBASIS + b] * Wrbf[b * 3 + c];
  escal[idx] = acc;
}

// CG tensor-product coupling for one edge chunk.
// block = 384 threads = one edge; thread t -> (path p, multiplicity u)
__global__ void coup_kernel(const float* __restrict__ srcf, const float* __restrict__ dstf,
                            const float* __restrict__ rsh, const float* __restrict__ escal,
                            const int* __restrict__ src_i, const int* __restrict__ dst_i,
                            const float* __restrict__ cg, float* __restrict__ coup, int e0) {
  const int e_local = blockIdx.x;
  const int e = e0 + e_local;
  int u = threadIdx.x, p = 0;
  while (u >= P_M[p]) { u -= P_M[p]; ++p; }
  const int l1 = P_L1[p], l2 = P_L2[p], l3 = P_L3[p];
  const int d1 = 2*l1 + 1, d2 = 2*l2 + 1, d3 = 2*l3 + 1;
  const int s = src_i[e], t = dst_i[e];
  // scaled edge spherical harmonic for l2
  float w[5];
  const float es = escal[(long)e * 3 + l2];
  for (int j = 0; j < d2; ++j) w[j] = rsh[(long)e * 9 + RSOFF_[l2] + j] * es;
  // x1 = srcf[src] + dstf[dst] for this (l1, u)
  float x[5];
  const int xb = OFFL_[l1] + u * d1;
  for (int i = 0; i < d1; ++i)
    x[i] = srcf[(long)s * NDIM + xb + i] + dstf[(long)t * NDIM + xb + i];
  const float* C = cg + CG_OFFP[p];
  const long ob = (long)e_local * 1184 + B3_[l3] + (long)(P_UO[p] + u) * d3;
  for (int k = 0; k < d3; ++k) {
    float acc = 0.0f;
    for (int i = 0; i < d1; ++i) {
      float m = 0.0f;
      for (int j = 0; j < d2; ++j) m += w[j] * C[(i * d2 + j) * d3 + k];
      acc += x[i] * m;
    }
    coup[ob + k] = acc;
  }
}

__device__ __forceinline__ unsigned ford(float f) {
  unsigned u = __float_as_uint(f);
  return (u & 0x80000000u) ? ~u : (u | 0x80000000u);
}
__device__ __forceinline__ float fdec(unsigned u) {
  unsigned b = (u & 0x80000000u) ? (u & 0x7fffffffu) : ~u;
  return __uint_as_float(b);
}

// attention logits: proj[e,h] = sum q[src]·key ; atomic-max into per-dst seg max
__global__ void proj_kernel(const float* __restrict__ q, const float* __restrict__ key,
                            const int* __restrict__ src_i, const int* __restrict__ dst_i,
                            float* __restrict__ logits, unsigned* __restrict__ smax, int e0) {
  int idx = blockIdx.x * blockDim.x + threadIdx.x;
  if (idx >= EC * NHEADS) return;
  const int e_local = idx >> 3, h = idx & 7;
  const int e = e0 + e_local;
  const int s = src_i[e];
  const float* qp = q + (long)s * 1920;
  const float* kp = key + (long)e_local * 1920;
  float acc = 0.0f;
  const int b0 = h * 64, b1 = 512 + h * 96, b2 = 1280 + h * 80;
  for (int j = 0; j < 64; ++j) acc += qp[b0 + j] * kp[b0 + j];   // l=0 (m*d=64)
  for (int j = 0; j < 96; ++j) acc += qp[b1 + j] * kp[b1 + j];   // l=1 (m*d=96)
  for (int j = 0; j < 80; ++j) acc += qp[b2 + j] * kp[b2 + j];   // l=2 (m*d=80)
  const float logit = acc * SCALE_F;
  logits[(long)e * NHEADS + h] = logit;
  atomicMax(&smax[(long)dst_i[e] * NHEADS + h], ford(logit));
}

__global__ void exp_kernel(float* __restrict__ logits, const unsigned* __restrict__ smax,
                           float* __restrict__ ssum, const int* __restrict__ dst_i) {
  int idx = blockIdx.x * blockDim.x + threadIdx.x;
  if (idx >= N_EDGES * NHEADS) return;
  const int e = idx >> 3, h = idx & 7;
  const int d = dst_i[e];
  const float mx = fdec(smax[(long)d * NHEADS + h]);
  const float ex = expf(logits[idx] - mx);
  logits[idx] = ex;
  atomicAdd(&ssum[(long)d * NHEADS + h], ex);
}

// attn-weighted scatter of values into msg (64 threads per edge)
__global__ void scatter_kernel(const float* __restrict__ ex, const float* __restrict__ ssum,
                               const int* __restrict__ dst_i, const float* __restrict__ val,
                               float* __restrict__ msg, int e0) {
  const int e_local = blockIdx.x * 4 + (threadIdx.x >> 6);
  const int t = threadIdx.x & 63;
  const int e = e0 + e_local;
  const int dn = dst_i[e];
  float att[NHEADS];
#pragma unroll
  for (int h = 0; h < NHEADS; ++h)
    att[h] = ex[(long)e * NHEADS + h] / ssum[(long)dn * NHEADS + h];
  const float* vp = val + (long)e_local * 1920;
  float* mp = msg + (long)dn * 1920;
  for (int j = t; j < 1920; j += 64) {
    int h;
    if (j < 512)       h = j >> 6;
    else if (j < 1280) h = (j - 512) / 96;
    else               h = (j - 1280) / 80;
    atomicAdd(&mp[j], att[h] * vp[j]);
  }
}

// invariants x0[n,u] = ||mb block||
__global__ void inv_kernel(const float* __restrict__ mb, float* __restrict__ x0) {
  int idx = blockIdx.x * blockDim.x + threadIdx.x;
  if (idx >= N_NODES * NIRR) return;
  const int n = idx / NIRR, u = idx % NIRR;
  const float* m = mb + (long)n * NDIM;
  float s;
  if (u < 64) {
    float v = m[u]; s = v * v;
  } else if (u < 96) {
    int b = 64 + (u - 64) * 3;
    s = m[b]*m[b] + m[b+1]*m[b+1] + m[b+2]*m[b+2];
  } else {
    int b = 160 + (u - 96) * 5;
    s = 0.0f;
#pragma unroll
    for (int i = 0; i < 5; ++i) { float v = m[b + i]; s += v * v; }
  }
  x0[idx] = sqrtf(s);
}

__global__ void ln_kernel(const float* __restrict__ x0, const float* __restrict__ g,
                          const float* __restrict__ b, float* __restrict__ x1n) {
  int n = blockIdx.x * blockDim.x + threadIdx.x;
  if (n >= N_NODES) return;
  const float* x = x0 + (long)n * NIRR;
  float mu = 0.0f;
  for (int u = 0; u < NIRR; ++u) mu += x[u];
  mu *= (1.0f / NIRR);
  float var = 0.0f;
  for (int u = 0; u < NIRR; ++u) { float d = x[u] - mu; var += d * d; }
  var *= (1.0f / NIRR);
  const float inv = rsqrtf(var + 1e-5f);
  float* o = x1n + (long)n * NIRR;
  for (int u = 0; u < NIRR; ++u) o[u] = (x[u] - mu) * inv * g[u] + b[u];
}

__global__ void act_kernel(float* __restrict__ h, const float* __restrict__ bias) {
  int idx = blockIdx.x * blockDim.x + threadIdx.x;
  if (idx >= N_NODES * NIRR) return;
  const float v = h[idx] + bias[idx % NIRR];
  h[idx] = v / (1.0f + expf(-v));        // silu
}

__global__ void out_kernel(const float* __restrict__ node, const float* __restrict__ mb,
                           const float* __restrict__ x0, const float* __restrict__ h2,
                           float* __restrict__ out) {
  int idx = blockIdx.x * blockDim.x + threadIdx.x;
  if (idx >= N_NODES * NDIM) return;
  const int n = idx / NDIM, j = idx % NDIM;
  const int u = (j < 64) ? j : (j < 160) ? 64 + (j - 64) / 3 : 96 + (j - 160) / 5;
  const float gsc = h2[(long)n * NIRR + u] / (x0[(long)n * NIRR + u] + 1e-6f);
  out[idx] = node[idx] + mb[idx] * gsc;
}

// =====================================================================
// host-side launcher
// =====================================================================
static void launch_gemm(hipStream_t s, const float* A, long sAm, long sAk,
                        const float* B, long sBk,
                        float* C, long sCm, long sCn, int M, int K, int N) {
  dim3 grid(M / 64, (N + 31) / 32, 1);   // M always a multiple of 64 here
  size_t shmem = (size_t)K * 32 * sizeof(float);
  if (sAk == 1)
    wmma_gemm_f32_t<true><<<grid, 128, shmem, s>>>(A, sAm, 1, B, sBk, C, sCm, sCn, K, N);
  else
    wmma_gemm_f32_t<false><<<grid, 128, shmem, s>>>(A, sAm, sAk, B, sBk, C, sCm, sCn, K, N);
}

extern "C" void kernel_launch(void* const* d_in, const int* in_sizes, int n_in,
                              void* d_out, int out_size, void* d_ws, size_t ws_size,
                              hipStream_t stream) {
  (void)in_sizes; (void)n_in; (void)out_size; (void)ws_size;
  const float* node  = (const float*)d_in[0];
  const float* rbf   = (const float*)d_in[1];
  const float* rsh   = (const float*)d_in[2];
  const int*   eidx  = (const int*)d_in[3];
  const float* Wrbf  = (const float*)d_in[4];
  const float* brbf  = (const float*)d_in[5];
  const float* ln_g  = (const float*)d_in[6];
  const float* ln_b  = (const float*)d_in[7];
  const float* W1    = (const float*)d_in[8];
  const float* b1    = (const float*)d_in[9];
  const float* W2    = (const float*)d_in[10];
  const float* b2    = (const float*)d_in[11];
  const float* Wq[3]   = {(const float*)d_in[12], (const float*)d_in[17], (const float*)d_in[22]};
  const float* Wsrc[3] = {(const float*)d_in[13], (const float*)d_in[18], (const float*)d_in[23]};
  const float* Wdst[3] = {(const float*)d_in[14], (const float*)d_in[19], (const float*)d_in[24]};
  const float* Wkv[3]  = {(const float*)d_in[15], (const float*)d_in[20], (const float*)d_in[25]};
  const float* Wmsg[3] = {(const float*)d_in[16], (const float*)d_in[21], (const float*)d_in[26]};
  const int* src_i = eidx;
  const int* dst_i = eidx + N_EDGES;
  float* outp = (float*)d_out;

  float* fws   = (float*)d_ws;
  float* cg    = fws + CG_WS;
  float* escal = fws + ESC_WS;
  float* srcf  = fws + SRCF_WS;
  float* dstf  = fws + DSTF_WS;
  float* qws   = fws + Q_WS;
  float* logit = fws + LOG_WS;
  unsigned* smax = (unsigned*)(fws + SMAX_WS);
  float* ssum  = fws + SSUM_WS;
  float* msg   = fws + MSG_WS;
  float* mb    = fws + MB_WS;
  float* x0    = fws + X0_WS;
  float* x1n   = fws + X1N_WS;
  float* h1    = fws + H1_WS;
  float* h2    = fws + H2_WS;
  float* coup  = fws + COUP_WS;
  float* kv    = fws + KV_WS;

  hipStream_t s = stream;

  // CG tables + zero the accumulators
  cg_gen_kernel<<<1, 32, 0, s>>>(cg);
  init_kernel<<<(N_NODES * 1920 + 255) / 256, 256, 0, s>>>(msg, smax, ssum);
  escal_kernel<<<(N_EDGES * 3 + 255) / 256, 256, 0, s>>>(rbf, Wrbf, brbf, escal);

  // node-side linears (WMMA): srcf, dstf, q   -- per-l, per m-component
  for (int l = 0; l < 3; ++l) {
    const int m = MUL_[l], d = DIM_[l];
    for (int i = 0; i < d; ++i) {
      launch_gemm(s, node + OFFL_[l] + i, NDIM, d, Wsrc[l], m,
                  srcf + OFFL_[l] + i, NDIM, d, N_NODES, m, m);
      launch_gemm(s, node + OFFL_[l] + i, NDIM, d, Wdst[l], m,
                  dstf + OFFL_[l] + i, NDIM, d, N_NODES, m, m);
      launch_gemm(s, node + OFFL_[l] + i, NDIM, d, Wq[l], NHEADS * m,
                  qws + QOFF_[l] + i, 1920, d, N_NODES, m, NHEADS * m);
    }
  }

  // pass 1 over edge chunks: coupling -> key GEMM -> logits + segment max
  for (int c = 0; c < NCHUNK; ++c) {
    const int e0 = c * EC;
    coup_kernel<<<EC, 384, 0, s>>>(srcf, dstf, rsh, escal, src_i, dst_i, cg, coup, e0);
    for (int l = 0; l < 3; ++l) {
      const int m = MUL_[l], d = DIM_[l];
      for (int i = 0; i < d; ++i)
        launch_gemm(s, coup + B3_[l] + i, 1184, d, Wkv[l], 2 * NHEADS * m,
                    kv + QOFF_[l] + i, 1920, d, EC, CM_[l], NHEADS * m);
    }
    proj_kernel<<<(EC * NHEADS + 255) / 256, 256, 0, s>>>(qws, kv, src_i, dst_i, logit, smax, e0);
  }

  // segment softmax normalization
  exp_kernel<<<(N_EDGES * NHEADS + 255) / 256, 256, 0, s>>>(logit, smax, ssum, dst_i);

  // pass 2 over edge chunks: recompute coupling -> value GEMM -> scatter
  for (int c = 0; c < NCHUNK; ++c) {
    const int e0 = c * EC;
    coup_kernel<<<EC, 384, 0, s>>>(srcf, dstf, rsh, escal, src_i, dst_i, cg, coup, e0);
    for (int l = 0; l < 3; ++l) {
      const int m = MUL_[l], d = DIM_[l];
      for (int i = 0; i < d; ++i)
        launch_gemm(s, coup + B3_[l] + i, 1184, d, Wkv[l] + NHEADS * m, 2 * NHEADS * m,
                    kv + QOFF_[l] + i, 1920, d, EC, CM_[l], NHEADS * m);
    }
    scatter_kernel<<<EC / 4, 256, 0, s>>>(logit, ssum, dst_i, kv, msg, e0);
  }

  // message linear: mb = msg @ Wmsg  (K = 8m)
  for (int l = 0; l < 3; ++l) {
    const int m = MUL_[l], d = DIM_[l];
    for (int i = 0; i < d; ++i)
      launch_gemm(s, msg + QOFF_[l] + i, 1920, d, Wmsg[l], m,
                  mb + OFFL_[l] + i, NDIM, d, N_NODES, NHEADS * m, m);
  }

  // NormGate: invariants -> LayerNorm -> MLP (WMMA) -> gate -> residual
  inv_kernel<<<(N_NODES * NIRR + 255) / 256, 256, 0, s>>>(mb, x0);
  ln_kernel<<<(N_NODES + 255) / 256, 256, 0, s>>>(x0, ln_g, ln_b, x1n);
  launch_gemm(s, x1n, NIRR, 1, W1, NIRR, h1, NIRR, 1, N_NODES, NIRR, NIRR);
  act_kernel<<<(N_NODES * NIRR + 255) / 256, 256, 0, s>>>(h1, b1);
  launch_gemm(s, h1, NIRR, 1, W2, NIRR, h2, NIRR, 1, N_NODES, NIRR, NIRR);
  act_kernel<<<(N_NODES * NIRR + 255) / 256, 256, 0, s>>>(h2, b2);
  out_kernel<<<(N_NODES * NDIM + 255) / 256, 256, 0, s>>>(node, mb, x0, h2, outp);
}